// PPYoloELoss_14542759264894
// MI455X (gfx1250) — compile-verified
//
#include <hip/hip_runtime.h>
#include <math.h>

#define B_   32
#define M_   100
#define A_   8400
#define NC_  80
#define MG_  7          // ceil(M/16) m-groups of 16
#define EPS_ 1e-9f

typedef __attribute__((ext_vector_type(2))) float v2f;
typedef __attribute__((ext_vector_type(8))) float v8f;

// ---------------------------------------------------------------- kernel 0
__global__ void k_init(int* __restrict__ cnt, int* __restrict__ mstar) {
  int i = blockIdx.x * blockDim.x + threadIdx.x;
  if (i < B_ * A_) { cnt[i] = 0; mstar[i] = 0x7fffffff; }
}

// ---------------------------------------------------------------- kernel 1
// One wave32 per (batch, group-of-16 GTs). WMMA f32 16x16x4 computes the
// GTcenter x AnchorCenter dot-product tile; dist^2 = |g|^2+|a|^2-2*dot.
// Lanes 0-15 each own one GT row and maintain a top-9 (smallest dist) list
// per pyramid level; then compute the 27 candidate IoUs, the mean+std
// threshold, the inside-gt test, and emit (anchor_idx, flag) candidates.
__global__ void __launch_bounds__(32) k_topk(
    const float* __restrict__ anchors,    // A*4
    const float* __restrict__ gt_boxes,   // B*M*4
    const float* __restrict__ pad_mask,   // B*M
    int*   __restrict__ cand_idx,         // B*M*27
    float* __restrict__ cand_flag)        // B*M*27
{
  __shared__ float s_gcx[16], s_gcy[16], s_gn[16];
  __shared__ float s_gb[16][4];
  __shared__ float s_an[16];
  __shared__ float s_tile[16][17];

  const int b    = blockIdx.x / MG_;
  const int mg   = blockIdx.x % MG_;
  const int lane = threadIdx.x;

  if (lane < 16) {
    int m = mg * 16 + lane;
    float4 g = (m < M_) ? ((const float4*)gt_boxes)[b * M_ + m]
                        : make_float4(0.f, 0.f, 0.f, 0.f);
    s_gb[lane][0] = g.x; s_gb[lane][1] = g.y;
    s_gb[lane][2] = g.z; s_gb[lane][3] = g.w;
    float cx = 0.5f * (g.x + g.z) - 320.0f;   // re-center to reduce cancellation
    float cy = 0.5f * (g.y + g.w) - 320.0f;
    s_gcx[lane] = cx; s_gcy[lane] = cy; s_gn[lane] = cx * cx + cy * cy;
  }
  __syncthreads();

  // A-operand (16x4, K=2..3 zero-padded): constant across anchor tiles.
  v2f Amat;
  Amat[0] = (lane < 16) ? s_gcx[lane] : 0.0f;
  Amat[1] = (lane < 16) ? s_gcy[lane] : 0.0f;

  int cidx[27];

  #pragma unroll 1
  for (int lv = 0; lv < 3; ++lv) {
    const int s0 = (lv == 0) ? 0    : (lv == 1) ? 6400 : 8000;
    const int sz = (lv == 0) ? 6400 : (lv == 1) ? 1600 : 400;

    float td[9]; int ti[9];
    #pragma unroll
    for (int q = 0; q < 9; ++q) { td[q] = 3.0e38f; ti[q] = s0; }
    float worst = 3.0e38f; int wp = 0;

    for (int n0 = s0; n0 < s0 + sz; n0 += 16) {
      float bx = 0.0f, by = 0.0f;
      if (lane < 16) {
        float4 ab = ((const float4*)anchors)[n0 + lane];
        bx = 0.5f * (ab.x + ab.z) - 320.0f;
        by = 0.5f * (ab.y + ab.w) - 320.0f;
        s_an[lane] = bx * bx + by * by;
      }
      __syncthreads();

      // B-operand (4x16): K rows 0,1 in lanes 0-15; K=2,3 zero in lanes 16-31.
      v2f Bmat; Bmat[0] = bx; Bmat[1] = by;
      v8f acc = {};
      acc = __builtin_amdgcn_wmma_f32_16x16x4_f32(false, Amat, false, Bmat,
                                                  (short)0, acc, false, false);

      const int   nl   = lane & 15;
      const float an   = s_an[nl];
      const int   mofs = (lane >= 16) ? 8 : 0;
      #pragma unroll
      for (int r = 0; r < 8; ++r) {
        float d2 = s_gn[r + mofs] + an - 2.0f * acc[r];
        s_tile[r + mofs][nl] = d2;
      }
      __syncthreads();

      if (lane < 16) {
        #pragma unroll
        for (int j = 0; j < 16; ++j) {
          float d = s_tile[lane][j];
          if (d < worst) {                  // strict <: keeps lowest index on ties
            td[wp] = d; ti[wp] = n0 + j;
            worst = td[0]; wp = 0;
            #pragma unroll
            for (int q = 1; q < 9; ++q)
              if (td[q] > worst) { worst = td[q]; wp = q; }
          }
        }
      }
      __syncthreads();
    }
    if (lane < 16) {
      #pragma unroll
      for (int q = 0; q < 9; ++q) cidx[lv * 9 + q] = ti[q];
    }
  }

  const int m = mg * 16 + lane;
  if (lane < 16 && m < M_) {
    const float pad = pad_mask[b * M_ + m];
    const float gx1 = s_gb[lane][0], gy1 = s_gb[lane][1];
    const float gx2 = s_gb[lane][2], gy2 = s_gb[lane][3];
    const float garea = fmaxf(gx2 - gx1, 0.f) * fmaxf(gy2 - gy1, 0.f);

    float g[27];
    #pragma unroll 1
    for (int k = 0; k < 27; ++k) {
      float4 ab = ((const float4*)anchors)[cidx[k]];
      float x1 = fmaxf(gx1, ab.x), y1 = fmaxf(gy1, ab.y);
      float x2 = fminf(gx2, ab.z), y2 = fminf(gy2, ab.w);
      float ov = fmaxf(x2 - x1, 0.f) * fmaxf(y2 - y1, 0.f);
      float aarea = fmaxf(ab.z - ab.x, 0.f) * fmaxf(ab.w - ab.y, 0.f);
      g[k] = (ov / (garea + aarea - ov + EPS_)) * pad;
    }
    float s = 0.f;
    #pragma unroll
    for (int k = 0; k < 27; ++k) s += g[k];
    const float mean = s * (1.0f / 27.0f);
    float v = 0.f;
    #pragma unroll
    for (int k = 0; k < 27; ++k) { float t = g[k] - mean; v += t * t; }
    const float thr = mean + sqrtf(v * (1.0f / 26.0f));   // std with ddof=1

    #pragma unroll 1
    for (int k = 0; k < 27; ++k) {
      float4 ab = ((const float4*)anchors)[cidx[k]];
      float acx = 0.5f * (ab.x + ab.z), acy = 0.5f * (ab.y + ab.w);
      float mind = fminf(fminf(acx - gx1, acy - gy1),
                         fminf(gx2 - acx, gy2 - acy));
      bool flag = (pad > 0.f) && (g[k] > thr) && (mind > EPS_);
      int o = (b * M_ + m) * 27 + k;
      cand_idx[o]  = cidx[k];
      cand_flag[o] = flag ? 1.0f : 0.0f;
    }
  }
}

// ---------------------------------------------------------------- kernel 2
__global__ void k_scatter(const int* __restrict__ cand_idx,
                          const float* __restrict__ cand_flag,
                          int* __restrict__ cnt, int* __restrict__ mstar) {
  int i = blockIdx.x * blockDim.x + threadIdx.x;
  if (i >= B_ * M_ * 27) return;
  if (cand_flag[i] > 0.f) {
    int b = i / (M_ * 27);
    int m = (i / 27) % M_;
    int a = cand_idx[i];
    atomicAdd(&cnt[b * A_ + a], 1);
    atomicMin(&mstar[b * A_ + a], m);   // lowest m == argmax of a 0/1 column
  }
}

// ---------------------------------------------------------------- kernel 3
__global__ void __launch_bounds__(256) k_final(
    const float* __restrict__ anchors,
    const int*   __restrict__ gt_labels,
    const float* __restrict__ gt_boxes,
    const float* __restrict__ pred_boxes,
    const int*   __restrict__ bgp,
    const int*   __restrict__ cnt,
    const int*   __restrict__ mstar,
    float* __restrict__ out_labels,
    float* __restrict__ out_bboxes,
    float* __restrict__ out_scores)
{
  __shared__ float s_gb[M_][4];
  __shared__ int   s_gl[M_];
  const int CH    = (A_ + 255) / 256;       // 33 anchor chunks per batch
  const int b     = blockIdx.x / CH;
  const int chunk = blockIdx.x % CH;
  const int t     = threadIdx.x;

  for (int i = t; i < M_; i += 256) {
    float4 g = ((const float4*)gt_boxes)[b * M_ + i];
    s_gb[i][0] = g.x; s_gb[i][1] = g.y; s_gb[i][2] = g.z; s_gb[i][3] = g.w;
    s_gl[i] = gt_labels[b * M_ + i];
  }
  __syncthreads();

  const int a = chunk * 256 + t;
  if (a >= A_) return;
  __builtin_prefetch(&pred_boxes[(size_t)(b * A_ + a) * 4], 0, 1);

  float4 ab = ((const float4*)anchors)[a];
  const float aarea = fmaxf(ab.z - ab.x, 0.f) * fmaxf(ab.w - ab.y, 0.f);

  // argmax over m of IoU(gt_m, anchor_a); first max wins (jnp.argmax).
  float best = -1.0f; int bm = 0;
  #pragma unroll 1
  for (int m = 0; m < M_; ++m) {
    float gx1 = s_gb[m][0], gy1 = s_gb[m][1], gx2 = s_gb[m][2], gy2 = s_gb[m][3];
    float x1 = fmaxf(gx1, ab.x), y1 = fmaxf(gy1, ab.y);
    float x2 = fminf(gx2, ab.z), y2 = fminf(gy2, ab.w);
    float ov = fmaxf(x2 - x1, 0.f) * fmaxf(y2 - y1, 0.f);
    float ga = fmaxf(gx2 - gx1, 0.f) * fmaxf(gy2 - gy1, 0.f);
    float iou = ov / (ga + aarea - ov + EPS_);
    if (iou > best) { best = iou; bm = m; }
  }

  const int c = cnt[b * A_ + a];
  int  agi; bool pos;
  if (c > 1)       { agi = bm;                 pos = true;  }
  else if (c == 1) { agi = mstar[b * A_ + a];  pos = true;  }
  else             { agi = 0;                  pos = false; }

  const int bg  = bgp[0];
  const int lab = s_gl[agi];
  out_labels[b * A_ + a] = pos ? (float)lab : (float)bg;

  float4 gbox = make_float4(s_gb[agi][0], s_gb[agi][1], s_gb[agi][2], s_gb[agi][3]);
  ((float4*)out_bboxes)[b * A_ + a] = gbox;

  float scale = 0.0f;
  if (pos) {
    float4 pb = ((const float4*)pred_boxes)[b * A_ + a];
    float x1 = fmaxf(gbox.x, pb.x), y1 = fmaxf(gbox.y, pb.y);
    float x2 = fminf(gbox.z, pb.z), y2 = fminf(gbox.w, pb.w);
    float ov = fmaxf(x2 - x1, 0.f) * fmaxf(y2 - y1, 0.f);
    float ga = fmaxf(gbox.z - gbox.x, 0.f) * fmaxf(gbox.w - gbox.y, 0.f);
    float pa = fmaxf(pb.z - pb.x, 0.f) * fmaxf(pb.w - pb.y, 0.f);
    scale = ov / (ga + pa - ov + EPS_);
  }

  float* sc = out_scores + (size_t)(b * A_ + a) * NC_;
  #pragma unroll 4
  for (int j = 0; j < NC_; ++j) sc[j] = 0.0f;
  if (pos && lab >= 0 && lab < NC_) sc[lab] = scale;
}

// ---------------------------------------------------------------- launch
extern "C" void kernel_launch(void* const* d_in, const int* in_sizes, int n_in,
                              void* d_out, int out_size, void* d_ws, size_t ws_size,
                              hipStream_t stream) {
  const float* anchors    = (const float*)d_in[0];
  const int*   gt_labels  = (const int*)  d_in[1];
  const float* gt_boxes   = (const float*)d_in[2];
  const float* pad_mask   = (const float*)d_in[3];
  const float* pred_boxes = (const float*)d_in[4];
  const int*   bgp        = (const int*)  d_in[5];

  float* out        = (float*)d_out;
  float* out_labels = out;                        // B*A
  float* out_bboxes = out + (size_t)B_ * A_;      // B*A*4
  float* out_scores = out + (size_t)B_ * A_ * 5;  // B*A*80

  char* w = (char*)d_ws;
  int*   cand_idx  = (int*)w;    w += (size_t)B_ * M_ * 27 * sizeof(int);
  float* cand_flag = (float*)w;  w += (size_t)B_ * M_ * 27 * sizeof(float);
  int*   cnt       = (int*)w;    w += (size_t)B_ * A_ * sizeof(int);
  int*   mstar     = (int*)w;

  k_init<<<(B_ * A_ + 255) / 256, 256, 0, stream>>>(cnt, mstar);
  k_topk<<<B_ * MG_, 32, 0, stream>>>(anchors, gt_boxes, pad_mask,
                                      cand_idx, cand_flag);
  k_scatter<<<(B_ * M_ * 27 + 255) / 256, 256, 0, stream>>>(cand_idx, cand_flag,
                                                            cnt, mstar);
  k_final<<<B_ * ((A_ + 255) / 256), 256, 0, stream>>>(
      anchors, gt_labels, gt_boxes, pred_boxes, bgp, cnt, mstar,
      out_labels, out_bboxes, out_scores);
}